// ShiTomasiScore_40587440947885
// MI455X (gfx1250) — compile-verified
//
#include <hip/hip_runtime.h>
#include <cstdint>

// Problem geometry (from reference setup_inputs)
#define IMG_HGT 1080
#define IMG_WID 1920

// Tile geometry
#define TILE_W 64
#define TILE_H 16
#define HALO_W 72             // staged cols [tx0-4, tx0+68): 16B-aligned float4 groups
#define HALO_H (TILE_H + 4)   // 20
#define NVEC   (HALO_W / 4)   // 18 b128 transfers per staged row
#define GRAD_W (TILE_W + 2)   // 66
#define GRAD_H (TILE_H + 2)   // 18
#define NTHREADS 256

__device__ __forceinline__ int clampi(int v, int lo, int hi) {
    return v < lo ? lo : (v > hi ? hi : v);
}

// CDNA5 async direct-to-LDS copies (ASYNCcnt-tracked, bypass VGPR round trip).
// Low 32 bits of a generic pointer to LDS are the wave-relative LDS byte
// offset (gfx1250 flat aperture mapping).
__device__ __forceinline__ void async_lds_b32(const float* lds_p, const float* gp) {
    asm volatile("global_load_async_to_lds_b32 %0, %1, off"
                 :
                 : "v"((uint32_t)(uintptr_t)lds_p), "v"((uint64_t)(uintptr_t)gp)
                 : "memory");
}
__device__ __forceinline__ void async_lds_b128(const float* lds_p, const float* gp) {
    asm volatile("global_load_async_to_lds_b128 %0, %1, off"
                 :
                 : "v"((uint32_t)(uintptr_t)lds_p), "v"((uint64_t)(uintptr_t)gp)
                 : "memory");
}

__global__ __launch_bounds__(NTHREADS, 4)
void shi_tomasi_kernel(const float* __restrict__ img, float* __restrict__ out,
                       int H, int W) {
    __shared__ float s_img[HALO_H * HALO_W];
    __shared__ float s_gxx[GRAD_H * GRAD_W];
    __shared__ float s_gyy[GRAD_H * GRAD_W];
    __shared__ float s_gxy[GRAD_H * GRAD_W];

    const int tid = threadIdx.x;
    const int tx0 = blockIdx.x * TILE_W;
    const int ty0 = blockIdx.y * TILE_H;
    const int b   = blockIdx.z;
    const float* imgb = img + (size_t)b * H * W;

    // ---------------- Phase 1: async halo staging into LDS ------------------
    // Fast path (28/30 tile columns): the x-halo [tx0-4, tx0+68) is fully
    // in-bounds, so rows are contiguous and 16B-aligned -> b128 transfers.
    // Y-clamping only changes the row address, so b128 stays valid at the
    // top/bottom image edges. Edge tile columns need per-element x-clamping
    // (replicate pad breaks contiguity) -> b32 fallback. Branch is uniform.
    const bool fastx = (tx0 >= 4) && (tx0 + HALO_W - 4 <= W);
    if (fastx) {
        for (int idx = tid; idx < HALO_H * NVEC; idx += NTHREADS) {
            const int r = idx / NVEC;
            const int j = idx - r * NVEC;
            const int gy = clampi(ty0 + r - 2, 0, H - 1);
            async_lds_b128(&s_img[r * HALO_W + 4 * j],
                           imgb + (size_t)gy * W + (tx0 - 4 + 4 * j));
        }
    } else {
        for (int idx = tid; idx < HALO_H * HALO_W; idx += NTHREADS) {
            const int r = idx / HALO_W;
            const int c = idx - r * HALO_W;
            const int gy = clampi(ty0 + r - 2, 0, H - 1);
            const int gx = clampi(tx0 + c - 4, 0, W - 1);
            async_lds_b32(&s_img[idx], imgb + (size_t)gy * W + gx);
        }
    }
    asm volatile("s_wait_asynccnt 0" ::: "memory");
    __syncthreads();

    // ---------------- Phase 2: gradients at CLAMPED positions ---------------
    // The reference box sum clamps the gradient index, so the gradient LDS
    // entry for an out-of-range position must hold the in-bounds gradient at
    // the clamped coordinate (NOT a gradient of replicated pixels).
    for (int idx = tid; idx < GRAD_H * GRAD_W; idx += NTHREADS) {
        const int r  = idx / GRAD_W;
        const int c  = idx - r * GRAD_W;
        const int gy = clampi(ty0 + r - 1, 0, H - 1);
        const int gx = clampi(tx0 + c - 1, 0, W - 1);
        // Translate the clamped global coordinate back into halo-LDS indices.
        const int ir = gy - ty0 + 2;   // in [1, HALO_H-2]
        const int ic = gx - tx0 + 4;   // in [2, HALO_W-3]
        const float* p = &s_img[ir * HALO_W + ic];
        const float tl = p[-HALO_W - 1], tc = p[-HALO_W], tr = p[-HALO_W + 1];
        const float ml = p[-1],                            mr = p[1];
        const float bl = p[ HALO_W - 1], bc = p[ HALO_W], br = p[ HALO_W + 1];
        const float ix = (tr + 2.0f * mr + br) - (tl + 2.0f * ml + bl);
        const float iy = (bl + 2.0f * bc + br) - (tl + 2.0f * tc + tr);
        s_gxx[idx] = ix * ix;
        s_gyy[idx] = iy * iy;
        s_gxy[idx] = ix * iy;
    }
    __syncthreads();

    // ---------------- Phase 3: 3x3 box sums + min eigenvalue ----------------
    #pragma unroll
    for (int k = 0; k < (TILE_H * TILE_W) / NTHREADS; ++k) {
        const int idx = tid + k * NTHREADS;
        const int r = idx / TILE_W;         // TILE_W = 64 -> shift
        const int c = idx - r * TILE_W;
        const int gy = ty0 + r;
        const int gx = tx0 + c;
        if (gy < H && gx < W) {
            float sxx = 0.0f, syy = 0.0f, sxy = 0.0f;
            #pragma unroll
            for (int dr = 0; dr < 3; ++dr) {
                const int base = (r + dr) * GRAD_W + c;
                sxx += s_gxx[base] + s_gxx[base + 1] + s_gxx[base + 2];
                syy += s_gyy[base] + s_gyy[base + 1] + s_gyy[base + 2];
                sxy += s_gxy[base] + s_gxy[base + 1] + s_gxy[base + 2];
            }
            const float half_trace = (sxx + syy) * 0.5f;
            const float diff_half  = (sxx - syy) * 0.5f;
            const float disc = __builtin_fmaf(diff_half, diff_half,
                               __builtin_fmaf(sxy, sxy, 1e-10f));
            const float lam = half_trace - __builtin_sqrtf(disc);
            out[((size_t)b * H + gy) * W + gx] = fmaxf(lam, 0.0f);
        }
    }
}

extern "C" void kernel_launch(void* const* d_in, const int* in_sizes, int n_in,
                              void* d_out, int out_size, void* d_ws, size_t ws_size,
                              hipStream_t stream) {
    const float* img = (const float*)d_in[0];
    float* out = (float*)d_out;
    const int H = IMG_HGT;
    const int W = IMG_WID;
    const int batch = in_sizes[0] / (H * W);   // 8 for the reference setup

    dim3 block(NTHREADS, 1, 1);
    dim3 grid((W + TILE_W - 1) / TILE_W,
              (H + TILE_H - 1) / TILE_H,
              batch);
    shi_tomasi_kernel<<<grid, block, 0, stream>>>(img, out, H, W);
}